// HKLinearDropIn_17523466567935
// MI455X (gfx1250) — compile-verified
//
#include <hip/hip_runtime.h>
#include <hip/hip_bf16.h>
#include <stdint.h>

typedef __attribute__((ext_vector_type(2))) float v2f;
typedef __attribute__((ext_vector_type(8))) float v8f;
typedef __attribute__((__vector_size__(4 * sizeof(int)))) int v4i;

#define IN_F 1024
#define OUT_F 4096
#define N_CLUSTERS 64
#define LN_EPS 1e-5f
#define THRESH 0.5f

#define AS1 __attribute__((address_space(1)))
#define AS3 __attribute__((address_space(3)))

#if __has_builtin(__builtin_amdgcn_global_load_async_to_lds_b128)
#define USE_ASYNC_LDS 1
#else
#define USE_ASYNC_LDS 0
#endif

#if __has_builtin(__builtin_amdgcn_s_wait_asynccnt)
#define WAIT_ASYNC(n) __builtin_amdgcn_s_wait_asynccnt(n)
#else
#define WAIT_ASYNC(n) asm volatile("s_wait_asynccnt %0" ::"i"(n) : "memory")
#endif

__device__ static inline AS1 v4i* to_global(const void* p) {
    return reinterpret_cast<AS1 v4i*>(reinterpret_cast<uintptr_t>(p));
}
__device__ static inline AS3 v4i* to_lds(void* p) {
    // generic LDS pointers carry the LDS byte offset in the low 32 bits
    return reinterpret_cast<AS3 v4i*>((unsigned int)reinterpret_cast<uintptr_t>(p));
}

// ---------------------------------------------------------------------------
// Kernel A: zero cluster-hit flags (workspace is poisoned, must init)
// ---------------------------------------------------------------------------
__global__ void zero_cmask_kernel(int* __restrict__ cmask) {
    if (threadIdx.x < N_CLUSTERS) cmask[threadIdx.x] = 0;
}

// ---------------------------------------------------------------------------
// Kernel B: per-row centroid dots -> layernorm -> softmax -> threshold
// one 64-thread block per row; thread c owns cluster c
// ---------------------------------------------------------------------------
__global__ __launch_bounds__(64) void key_search_kernel(
    const float* __restrict__ x,         // [Nrows, IN_F]
    const float* __restrict__ centroids, // [IN_F, N_CLUSTERS]
    const float* __restrict__ lnw,       // [N_CLUSTERS]
    float* __restrict__ qmaskf,          // [Nrows] out: 1.0 / 0.0
    int* __restrict__ cmask)             // [N_CLUSTERS] hit flags (atomicOr)
{
    __shared__ float xs[IN_F];
    __shared__ float red[N_CLUSTERS];
    __shared__ int anyflag;

    const int tid = threadIdx.x;          // 0..63 == cluster id
    const long long row = blockIdx.x;

    const float* xrow = x + row * IN_F;
#pragma unroll
    for (int i = 0; i < IN_F / N_CLUSTERS; ++i)
        xs[i * N_CLUSTERS + tid] = xrow[i * N_CLUSTERS + tid];
    if (tid == 0) anyflag = 0;
    __syncthreads();

    float s = 0.0f;
    const float* cp = centroids + tid;
#pragma unroll 8
    for (int k = 0; k < IN_F; ++k)
        s += xs[k] * cp[(size_t)k * N_CLUSTERS];

    // mean / var over 64 clusters
    red[tid] = s; __syncthreads();
    for (int off = 32; off > 0; off >>= 1) {
        if (tid < off) red[tid] += red[tid + off];
        __syncthreads();
    }
    const float mean = red[0] * (1.0f / N_CLUSTERS);
    __syncthreads();

    const float d = s - mean;
    red[tid] = d * d; __syncthreads();
    for (int off = 32; off > 0; off >>= 1) {
        if (tid < off) red[tid] += red[tid + off];
        __syncthreads();
    }
    const float var = red[0] * (1.0f / N_CLUSTERS);
    __syncthreads();

    const float dn = d * rsqrtf(var + LN_EPS) * lnw[tid];

    // softmax: max then exp-sum
    red[tid] = dn; __syncthreads();
    for (int off = 32; off > 0; off >>= 1) {
        if (tid < off) red[tid] = fmaxf(red[tid], red[tid + off]);
        __syncthreads();
    }
    const float mx = red[0];
    __syncthreads();

    const float e = expf(dn - mx);
    red[tid] = e; __syncthreads();
    for (int off = 32; off > 0; off >>= 1) {
        if (tid < off) red[tid] += red[tid + off];
        __syncthreads();
    }
    const float esum = red[0];

    const float prob = e / esum;
    const bool hit = prob > THRESH;
    if (hit) {
        atomicOr(&anyflag, 1);
        atomicOr(&cmask[tid], 1);
    }
    __syncthreads();
    if (tid == 0) qmaskf[row] = anyflag ? 1.0f : 0.0f;
}

// ---------------------------------------------------------------------------
// Kernel C: key mask from cluster assignment (int64 indices)
// ---------------------------------------------------------------------------
__global__ void build_kmask_kernel(const int* __restrict__ cmask,
                                   const long long* __restrict__ cassign,
                                   float* __restrict__ kmaskf) {
    const int o = blockIdx.x * blockDim.x + threadIdx.x;
    if (o < OUT_F) {
        const int c = (int)cassign[o];
        kmaskf[o] = cmask[c] ? 1.0f : 0.0f;
    }
}

// ---------------------------------------------------------------------------
// Kernel D: masked GEMM  out = (x @ W.T + bias) * (qmask x kmask)
// fp32 WMMA 16x16x4, 128x128 block tile, 8 waves each 64x32,
// double-buffered LDS fed by GLOBAL_LOAD_ASYNC_TO_LDS_B128 (ASYNCcnt)
// ---------------------------------------------------------------------------
#define BM 128
#define BN 128
#define BK 16
#define LDST 20           // padded LDS row stride (16B aligned, conflict-free)
#define TILE_F (BM * LDST)

__global__ __launch_bounds__(256) void gemm_masked_kernel(
    const float* __restrict__ X,      // [Nrows, IN_F]
    const float* __restrict__ W,      // [OUT_F, IN_F]  (B matrix, K-contiguous)
    const float* __restrict__ bias,   // [OUT_F]
    const float* __restrict__ qmaskf, // [Nrows]
    const float* __restrict__ kmaskf, // [OUT_F]
    float* __restrict__ out)          // [Nrows, OUT_F]
{
    __shared__ float As[2 * TILE_F];
    __shared__ float Bs[2 * TILE_F];
    __shared__ float qsh[BM];
    __shared__ float ksh[BN];

    const int tid   = threadIdx.x;
    const int lane  = tid & 31;
    const int wave  = tid >> 5;       // 0..7
    const int waveM = wave & 1;       // 2 wave-rows of 64
    const int waveN = wave >> 1;      // 4 wave-cols of 32
    const int m0 = blockIdx.y * BM;
    const int n0 = blockIdx.x * BN;

    if (tid < BM)            qsh[tid]      = qmaskf[m0 + tid];
    else if (tid < BM + BN)  ksh[tid - BM] = kmaskf[n0 + (tid - BM)];

    // per-thread copy slice: 2 float4 for A + 2 for B per tile
    const int e0   = tid;             // 0..255
    const int e1   = tid + 256;       // 256..511
    const int rowA0 = e0 >> 2, c40 = (e0 & 3) << 2;
    const int rowA1 = e1 >> 2, c41 = (e1 & 3) << 2;

    auto issue_tile = [&](int k0, int buf) {
        float* Ab = As + buf * TILE_F;
        float* Bb = Bs + buf * TILE_F;
#if USE_ASYNC_LDS
        __builtin_amdgcn_global_load_async_to_lds_b128(
            to_global(X + (size_t)(m0 + rowA0) * IN_F + k0 + c40),
            to_lds(&Ab[rowA0 * LDST + c40]), 0, 0);
        __builtin_amdgcn_global_load_async_to_lds_b128(
            to_global(X + (size_t)(m0 + rowA1) * IN_F + k0 + c41),
            to_lds(&Ab[rowA1 * LDST + c41]), 0, 0);
        __builtin_amdgcn_global_load_async_to_lds_b128(
            to_global(W + (size_t)(n0 + rowA0) * IN_F + k0 + c40),
            to_lds(&Bb[rowA0 * LDST + c40]), 0, 0);
        __builtin_amdgcn_global_load_async_to_lds_b128(
            to_global(W + (size_t)(n0 + rowA1) * IN_F + k0 + c41),
            to_lds(&Bb[rowA1 * LDST + c41]), 0, 0);
#else
        *(float4*)(&Ab[rowA0 * LDST + c40]) =
            *(const float4*)(X + (size_t)(m0 + rowA0) * IN_F + k0 + c40);
        *(float4*)(&Ab[rowA1 * LDST + c41]) =
            *(const float4*)(X + (size_t)(m0 + rowA1) * IN_F + k0 + c41);
        *(float4*)(&Bb[rowA0 * LDST + c40]) =
            *(const float4*)(W + (size_t)(n0 + rowA0) * IN_F + k0 + c40);
        *(float4*)(&Bb[rowA1 * LDST + c41]) =
            *(const float4*)(W + (size_t)(n0 + rowA1) * IN_F + k0 + c41);
#endif
    };

    const v8f vzero = {0.f, 0.f, 0.f, 0.f, 0.f, 0.f, 0.f, 0.f};
    v8f acc[4][2];
#pragma unroll
    for (int i = 0; i < 4; ++i)
#pragma unroll
        for (int j = 0; j < 2; ++j) acc[i][j] = vzero;

    const int lo16 = lane & 15;
    const int hi   = lane >> 4;       // 0 or 1

    const int nTiles = IN_F / BK;     // 64
    issue_tile(0, 0);                 // prologue: tile 0 -> buffer 0

    for (int t = 0; t < nTiles; ++t) {
        const int cur = t & 1;
        if (t + 1 < nTiles) {
            issue_tile((t + 1) * BK, cur ^ 1);   // prefetch next tile
#if USE_ASYNC_LDS
            WAIT_ASYNC(4);   // in-order: only the 4 just-issued remain => tile t done
#endif
        } else {
#if USE_ASYNC_LDS
            WAIT_ASYNC(0);
#endif
        }
        __syncthreads();     // all waves' tile-t deposits visible

        const float* Ab = As + cur * TILE_F;
        const float* Bb = Bs + cur * TILE_F;
#pragma unroll
        for (int kk = 0; kk < BK; kk += 4) {
            const int kh = kk + (hi << 1);  // lanes 0-15: k,k+1; lanes 16-31: k+2,k+3
            v2f a[4], b[2];
#pragma unroll
            for (int i = 0; i < 4; ++i) {
                const int r = waveM * 64 + i * 16 + lo16;
                a[i] = *(const v2f*)(&Ab[r * LDST + kh]);
            }
#pragma unroll
            for (int j = 0; j < 2; ++j) {
                const int c = waveN * 32 + j * 16 + lo16;
                b[j] = *(const v2f*)(&Bb[c * LDST + kh]);
            }
#pragma unroll
            for (int i = 0; i < 4; ++i)
#pragma unroll
                for (int j = 0; j < 2; ++j)
                    acc[i][j] = __builtin_amdgcn_wmma_f32_16x16x4_f32(
                        false, a[i], false, b[j], (short)0, acc[i][j], false, false);
        }
        __syncthreads();     // done reading buf `cur` before it is refilled
    }

    // epilogue: D layout -> lane holds col = lo16, rows r + 8*hi per VGPR r
#pragma unroll
    for (int j = 0; j < 2; ++j) {
        const int c  = waveN * 32 + j * 16 + lo16;
        const float kf = ksh[c];
        const float bv = bias[n0 + c];
#pragma unroll
        for (int i = 0; i < 4; ++i) {
#pragma unroll
            for (int r = 0; r < 8; ++r) {
                const int row = waveM * 64 + i * 16 + (hi << 3) + r;
                const float v = (acc[i][j][r] + bv) * (qsh[row] * kf);
                out[(size_t)(m0 + row) * OUT_F + (n0 + c)] = v;
            }
        }
    }
}

// ---------------------------------------------------------------------------
extern "C" void kernel_launch(void* const* d_in, const int* in_sizes, int n_in,
                              void* d_out, int out_size, void* d_ws, size_t ws_size,
                              hipStream_t stream) {
    const float*     x        = (const float*)d_in[0];
    const float*     weight   = (const float*)d_in[1];
    const float*     bias     = (const float*)d_in[2];
    const float*     centroid = (const float*)d_in[3];
    const float*     lnw      = (const float*)d_in[4];
    const long long* cassign  = (const long long*)d_in[5];
    float* out = (float*)d_out;

    const int Nrows = in_sizes[0] / IN_F;   // 4*4096 = 16384

    int*   cmask  = (int*)d_ws;                         // 64 ints
    float* qmaskf = (float*)((char*)d_ws + 256);        // [Nrows]
    float* kmaskf = qmaskf + Nrows;                     // [OUT_F]

    zero_cmask_kernel<<<1, 64, 0, stream>>>(cmask);
    key_search_kernel<<<Nrows, 64, 0, stream>>>(x, centroid, lnw, qmaskf, cmask);
    build_kmask_kernel<<<(OUT_F + 255) / 256, 256, 0, stream>>>(cmask, cassign, kmaskf);

    dim3 grid(OUT_F / BN, Nrows / BM);   // (32, 128)
    gemm_masked_kernel<<<grid, 256, 0, stream>>>(x, weight, bias, qmaskf, kmaskf, out);
}